// ScatteringFilter__42356967473550
// MI455X (gfx1250) — compile-verified
//
#include <hip/hip_runtime.h>

// ScatteringFilter for MI455X (gfx1250, wave32).
// out[r,d,f] = sum_{i: rows[i]==r} v_i * LTx[cols[i],d] * w_wav[cols[i],f]
// LTx[c,d]   = sum_{i: cols[i]==c} v_i * x[rows[i],d]
// Entire working set (~86 MB) is resident in the 192 MB L2 -> the kernel is
// L2 atomic-throughput bound; we fuse the outer product into the final
// scatter to cut the per-entry gather from 512 B to 96 B.

#define D_DIM 16
#define F_DIM 8

// ---------- gfx1250 async global->LDS staging (guarded for portability) ----
#if defined(__gfx1250__) && __has_builtin(__builtin_amdgcn_global_load_async_to_lds_b32)
#define HAVE_ASYNC_LDS 1
#else
#define HAVE_ASYNC_LDS 0
#endif

#if HAVE_ASYNC_LDS
typedef __attribute__((address_space(1))) int g_int;  // global addrspace
typedef __attribute__((address_space(3))) int l_int;  // LDS addrspace

__device__ __forceinline__ void async_copy_b32(const void* g, void* l) {
  __builtin_amdgcn_global_load_async_to_lds_b32(
      (g_int*)g, (l_int*)l, /*offset=*/0, /*cpol=*/0);
}
__device__ __forceinline__ void wait_async0() {
#if __has_builtin(__builtin_amdgcn_s_wait_asynccnt)
  __builtin_amdgcn_s_wait_asynccnt(0);
#else
  asm volatile("s_wait_asynccnt 0x0" ::: "memory");
#endif
}
#endif

// ---------------------------------------------------------------------------
// Pass 2 (dominant; placed FIRST in the module so the disasm snippet shows
// it): out[r, d, f] += v * LTx[c, d] * w_wav[c, f].
// One wave per COO entry; lane L owns (d = L/2, filters (L&1)*4 .. +3), so a
// wave emits the full 16x8 tile with 4 independent f32 atomics per lane.
// Each wave async-stages its own 32 entry triples into LDS (ASYNCcnt path,
// no cross-wave dependency -> no workgroup barrier needed).
__global__ void sf_scatter_out_kernel(const int* __restrict__ rows,
                                      const int* __restrict__ cols,
                                      const float* __restrict__ vals,
                                      const float* __restrict__ ltx,
                                      const float* __restrict__ wwav,
                                      float* __restrict__ out, long long nnz) {
  __shared__ int   sR[256];
  __shared__ int   sC[256];
  __shared__ float sV[256];

  const int lane = threadIdx.x & 31;
  const int wave = threadIdx.x >> 5;
  const int slot0 = wave * 32;
  const long long waveBase = (long long)blockIdx.x * 256 + slot0;

  int count = 0;
  if (waveBase < nnz) {
    long long rem = nnz - waveBase;
    count = rem < 32 ? (int)rem : 32;
  }

#if HAVE_ASYNC_LDS
  if (lane < count) {
    long long gi = waveBase + lane;
    async_copy_b32(rows + gi, &sR[slot0 + lane]);
    async_copy_b32(cols + gi, &sC[slot0 + lane]);
    async_copy_b32(vals + gi, &sV[slot0 + lane]);
  }
  wait_async0();
#else
  if (lane < count) {
    long long gi = waveBase + lane;
    sR[slot0 + lane] = rows[gi];
    sC[slot0 + lane] = cols[gi];
    sV[slot0 + lane] = vals[gi];
  }
#endif
  __builtin_amdgcn_wave_barrier();  // keep LDS stage before LDS reads

  const int d  = lane >> 1;
  const int fb = (lane & 1) * 4;

  for (int e = 0; e < count; ++e) {
    int   r = sR[slot0 + e];
    int   c = sC[slot0 + e];
    float v = sV[slot0 + e];

    // Warm the next entry's gather rows in L2/L0 while this entry's atomics
    // drain (global_prefetch_b8).
    if (lane < 2 && e + 1 < count) {
      int cn = sC[slot0 + e + 1];
      const float* pf = lane ? (wwav + (size_t)cn * F_DIM)
                             : (ltx + (size_t)cn * D_DIM);
      __builtin_prefetch(pf, 0, 0);
    }

    float a = v * ltx[(size_t)c * D_DIM + d];                 // coalesced 64 B row
    const float4 wv = *(const float4*)(wwav + (size_t)c * F_DIM + fb);  // broadcast 32 B row

    float* o = out + (size_t)r * (D_DIM * F_DIM) + d * F_DIM + fb;
    atomicAdd(o + 0, a * wv.x);
    atomicAdd(o + 1, a * wv.y);
    atomicAdd(o + 2, a * wv.z);
    atomicAdd(o + 3, a * wv.w);
  }
}

// ---------------------------------------------------------------------------
// Wavelet weights: w_wav[n,k] = e^{-eig}^{2^k} - e^{-eig}^{2^{k+1}}, k=0..7
// Repeated squaring: p_{k+1} = p_k^2, p_0 = exp(-eig).
__global__ void sf_wavelet_kernel(const float* __restrict__ eig,
                                  float* __restrict__ wwav, int n) {
  int i = blockIdx.x * blockDim.x + threadIdx.x;
  if (i >= n) return;
  float w[F_DIM + 1];
  float p = expf(-eig[i]);
  w[0] = p;
#pragma unroll
  for (int k = 1; k <= F_DIM; ++k) { p = p * p; w[k] = p; }
  float4 a = make_float4(w[0] - w[1], w[1] - w[2], w[2] - w[3], w[3] - w[4]);
  float4 b = make_float4(w[4] - w[5], w[5] - w[6], w[6] - w[7], w[7] - w[8]);
  float4* dst = (float4*)(wwav + (size_t)i * F_DIM);
  dst[0] = a;
  dst[1] = b;
}

// ---------------------------------------------------------------------------
// Pass 1: LTx[c, d] += v * x[r, d].  One thread per (entry, d) pair:
// 16 lanes share one entry -> x-row gather coalesces to one 64 B line,
// index loads broadcast within the half-wave.
__global__ void sf_scatter_ltx_kernel(const float* __restrict__ x,
                                      const int* __restrict__ rows,
                                      const int* __restrict__ cols,
                                      const float* __restrict__ vals,
                                      float* __restrict__ ltx, long long nnz) {
  long long t = (long long)blockIdx.x * blockDim.x + threadIdx.x;
  if (t >= nnz * D_DIM) return;
  long long i = t >> 4;
  int d = (int)(t & (D_DIM - 1));
  int r = rows[i];
  int c = cols[i];
  float v = vals[i];
  atomicAdd(&ltx[(size_t)c * D_DIM + d], v * x[(size_t)r * D_DIM + d]);
}

// ---------------------------------------------------------------------------
extern "C" void kernel_launch(void* const* d_in, const int* in_sizes, int n_in,
                              void* d_out, int out_size, void* d_ws, size_t ws_size,
                              hipStream_t stream) {
  const float* x    = (const float*)d_in[0];
  const int*   rows = (const int*)d_in[1];
  const int*   cols = (const int*)d_in[2];
  const float* vals = (const float*)d_in[3];
  const float* eig  = (const float*)d_in[4];

  const int       N   = in_sizes[0] / D_DIM;
  const long long nnz = in_sizes[1];

  float* out  = (float*)d_out;
  float* ltx  = (float*)d_ws;                                       // [N,16]  6.4 MB
  float* wwav = (float*)((char*)d_ws + (size_t)N * D_DIM * sizeof(float)); // [N,8] 3.2 MB

  // Accumulators must start at zero every call (harness poisons buffers).
  (void)hipMemsetAsync(out, 0, (size_t)N * D_DIM * F_DIM * sizeof(float), stream);
  (void)hipMemsetAsync(ltx, 0, (size_t)N * D_DIM * sizeof(float), stream);

  sf_wavelet_kernel<<<(N + 255) / 256, 256, 0, stream>>>(eig, wwav, N);

  long long tot1 = nnz * D_DIM;
  sf_scatter_ltx_kernel<<<(unsigned)((tot1 + 255) / 256), 256, 0, stream>>>(
      x, rows, cols, vals, ltx, nnz);

  sf_scatter_out_kernel<<<(unsigned)((nnz + 255) / 256), 256, 0, stream>>>(
      rows, cols, vals, ltx, wwav, out, nnz);
}